// RNN_71141838290984
// MI455X (gfx1250) — compile-verified
//
#include <hip/hip_runtime.h>
#include <hip/hip_bf16.h>

typedef __attribute__((ext_vector_type(2))) float v2f;
typedef __attribute__((ext_vector_type(4))) float v4f;
typedef __attribute__((ext_vector_type(8))) float v8f;

#define SEQ  256
#define BAT  16
#define VOC  32000
#define EMB  32
#define HID  16
#define ROWS (SEQ * BAT)   // 4096

// ---------------------------------------------------------------------------
// Kernel 1: ax[r][h] = b_x[h] + sum_e emb[x[r]][e] * W_x[e][h]
// Trivial FLOPs; plain VALU.
// ---------------------------------------------------------------------------
__global__ void rnn_ax_kernel(const int* __restrict__ x,
                              const float* __restrict__ emb,
                              const float* __restrict__ Wx,
                              const float* __restrict__ bx,
                              float* __restrict__ ax) {
    int idx = blockIdx.x * blockDim.x + threadIdx.x;   // 0..65535
    int r = idx >> 4;
    int h = idx & 15;
    int tok = x[r];
    const float* e = emb + (size_t)tok * EMB;
    float acc = bx[h];
#pragma unroll
    for (int k = 0; k < EMB; ++k)
        acc = fmaf(e[k], Wx[k * HID + h], acc);
    ax[idx] = acc;
}

// ---------------------------------------------------------------------------
// Kernel 2: sequential scan  h_t = sigmoid(ax_t + h_{t-1} @ W_h + b_h)
// State is 16x16; one workgroup, thread (b,j) owns one state element,
// W_h column j held in registers, state in LDS. Latency-bound by design.
// ---------------------------------------------------------------------------
__global__ void rnn_scan_kernel(const float* __restrict__ ax,
                                const float* __restrict__ Wh,
                                const float* __restrict__ bh,
                                float* __restrict__ hout) {
    __shared__ float hs[BAT][HID];
    int tid = threadIdx.x;          // 0..255
    int b = tid >> 4;
    int j = tid & 15;
    float wcol[HID];
#pragma unroll
    for (int k = 0; k < HID; ++k) wcol[k] = Wh[k * HID + j];
    float bj = bh[j];
    hs[b][j] = 0.0f;
    __syncthreads();
    for (int t = 0; t < SEQ; ++t) {
        float acc = ax[t * (BAT * HID) + tid] + bj;
#pragma unroll
        for (int k = 0; k < HID; ++k)
            acc = fmaf(hs[b][k], wcol[k], acc);
        float hn = 1.0f / (1.0f + __expf(-acc));
        __syncthreads();                 // all reads of old state done
        hs[b][j] = hn;
        hout[t * (BAT * HID) + tid] = hn;
        __syncthreads();                 // new state visible
    }
}

// ---------------------------------------------------------------------------
// Kernel 3: logits = H_tile(16x16) @ W_out(16x32000) via V_WMMA_F32_16X16X4_F32
// (K=16 as 4 chained K=4 WMMAs), fused online logsumexp in registers, then an
// in-place b128 normalization pass over the wave's own write-set with
// non-temporal final stores.
// One block per 16-row tile (= one timestep), 8 waves x 4000 vocab cols each.
// ---------------------------------------------------------------------------
__global__ void __launch_bounds__(256, 1)
rnn_logits_lsm_kernel(const float* __restrict__ hmat,   // [ROWS][HID]
                      const float* __restrict__ Wout,   // [HID][VOC]
                      float* __restrict__ out) {        // [ROWS][VOC]
    __shared__ float part_m[8][16];
    __shared__ float part_s[8][16];
    __shared__ float lse[16];

    const int tile = blockIdx.x;        // 0..255
    const int tid  = threadIdx.x;
    const int wave = tid >> 5;          // 0..7
    const int lane = tid & 31;
    const int lh   = lane & 15;         // N (or M for A-frag)
    const int hi   = lane >> 4;         // half-wave select

    const float* Ht = hmat + (size_t)tile * (16 * HID);

    // A fragments (ISA 16x4 f32 layout: lane M = lane%16,
    // VGPR0 holds K=0 / K=2 per half, VGPR1 holds K=1 / K=3)
    v2f afrag[4];
#pragma unroll
    for (int k4 = 0; k4 < 4; ++k4) {
        int kb = 4 * k4 + 2 * hi;
        afrag[k4].x = Ht[lh * HID + kb + 0];
        afrag[k4].y = Ht[lh * HID + kb + 1];
    }

    const int vbeg = wave * (VOC / 8);  // 4000 columns per wave
    const int vend = vbeg + (VOC / 8);

    float rm[8], rs[8];
#pragma unroll
    for (int i = 0; i < 8; ++i) { rm[i] = -__builtin_huge_valf(); rs[i] = 0.0f; }

    float* orow = out + (size_t)tile * 16 * VOC;

    for (int v0 = vbeg; v0 < vend; v0 += 16) {
        // prefetch B stream ahead (gfx1250 global_prefetch path)
        if (v0 + 64 < vend)
            __builtin_prefetch(&Wout[(size_t)(2 * hi) * VOC + v0 + 64 + lh], 0, 1);

        v8f c = {0.f, 0.f, 0.f, 0.f, 0.f, 0.f, 0.f, 0.f};
#pragma unroll
        for (int k4 = 0; k4 < 4; ++k4) {
            int kr = 4 * k4 + 2 * hi;
            v2f bfrag;
            bfrag.x = Wout[(size_t)(kr + 0) * VOC + v0 + lh];
            bfrag.y = Wout[(size_t)(kr + 1) * VOC + v0 + lh];
            // D = A*B + C ; exact f32 path: V_WMMA_F32_16X16X4_F32
            c = __builtin_amdgcn_wmma_f32_16x16x4_f32(
                    false, afrag[k4], false, bfrag, (short)0, c, false, false);
        }
        // store logits + online logsumexp (C/D layout: VGPR i -> row hi*8+i)
#pragma unroll
        for (int i = 0; i < 8; ++i) {
            float val = c[i];
            int m = hi * 8 + i;
            orow[(size_t)m * VOC + v0 + lh] = val;
            float mo = rm[i];
            float mn = fmaxf(mo, val);
            rs[i] = rs[i] * __expf(mo - mn) + __expf(val - mn);
            rm[i] = mn;
        }
    }

    // reduce (max,sum) across the 16 lanes of each half-wave (N dimension)
#pragma unroll
    for (int off = 1; off < 16; off <<= 1) {
#pragma unroll
        for (int i = 0; i < 8; ++i) {
            float om = __shfl_xor(rm[i], off, 32);
            float os = __shfl_xor(rs[i], off, 32);
            float mn = fmaxf(rm[i], om);
            rs[i] = rs[i] * __expf(rm[i] - mn) + os * __expf(om - mn);
            rm[i] = mn;
        }
    }
    if (lh == 0) {
#pragma unroll
        for (int i = 0; i < 8; ++i) {
            part_m[wave][hi * 8 + i] = rm[i];
            part_s[wave][hi * 8 + i] = rs[i];
        }
    }
    __syncthreads();
    if (tid < 16) {
        float m = part_m[0][tid], s = part_s[0][tid];
#pragma unroll
        for (int w = 1; w < 8; ++w) {
            float om = part_m[w][tid], os = part_s[w][tid];
            float mn = fmaxf(m, om);
            s = s * __expf(m - mn) + os * __expf(om - mn);
            m = mn;
        }
        lse[tid] = m + __logf(s);
    }
    __syncthreads();

    // Ensure this wave's pass-1 stores have landed before re-reading them
    // (same-wave store->load ordering; explicit wait for safety).
    asm volatile("s_wait_storecnt 0x0" ::: "memory");

    // pass 2: wave's write-set == wave's read-set (rows 0..15, cols
    // [vbeg,vend)). Stream it back row-major as b128 and normalize; final
    // stores are non-temporal (never re-read -> keep L2 for Wout/logits).
    for (int m = 0; m < 16; ++m) {
        float ls = lse[m];
        v4f* rowp = (v4f*)(orow + (size_t)m * VOC + vbeg);
        for (int j = lane; j < (VOC / 8) / 4; j += 32) {   // 1000 v4f/row
            v4f v = rowp[j];
            v.x -= ls; v.y -= ls; v.z -= ls; v.w -= ls;
            __builtin_nontemporal_store(v, &rowp[j]);
        }
    }
}

// ---------------------------------------------------------------------------
extern "C" void kernel_launch(void* const* d_in, const int* in_sizes, int n_in,
                              void* d_out, int out_size, void* d_ws, size_t ws_size,
                              hipStream_t stream) {
    const int*   x    = (const int*)d_in[0];
    const float* emb  = (const float*)d_in[1];
    const float* Wx   = (const float*)d_in[2];
    const float* bx   = (const float*)d_in[3];
    const float* Wh   = (const float*)d_in[4];
    const float* bh   = (const float*)d_in[5];
    const float* Wout = (const float*)d_in[6];
    float* out = (float*)d_out;

    float* ws_ax = (float*)d_ws;            // ROWS*HID floats (256 KB)
    float* ws_h  = ws_ax + ROWS * HID;      // ROWS*HID floats (256 KB)

    rnn_ax_kernel<<<(ROWS * HID) / 256, 256, 0, stream>>>(x, emb, Wx, bx, ws_ax);
    rnn_scan_kernel<<<1, 256, 0, stream>>>(ws_ax, Wh, bh, ws_h);
    rnn_logits_lsm_kernel<<<ROWS / 16, 256, 0, stream>>>(ws_h, Wout, out);
}